// DerivativeRBF_19816979103948
// MI455X (gfx1250) — compile-verified
//
#include <hip/hip_runtime.h>
#include <hip/hip_bf16.h>

typedef __attribute__((ext_vector_type(2))) float v2f;
typedef __attribute__((ext_vector_type(8))) float v8f;

// ---------------------------------------------------------------------------
// Sizes (fixed by the reference): N = M = 512, D = 16.
// d_out layout: K (512*512) | grad_K (8192*512) | hess_K (8192*8192)
// ---------------------------------------------------------------------------
#define NPTS 512
#define DDIM 16

// workspace float offsets
#define WS_XS    0         // Xs  = X / ls          (512 x 16, row-major)
#define WS_X2S   8192      // X2s = X2 / ls         (512 x 16, row-major)
#define WS_PC    16384     // Pc  = X  * ls^-2      (16 x 512, column-major)
#define WS_PC2   24576     // Pc2 = X2 * ls^-2      (16 x 512, column-major)
#define WS_NX    32768     // |Xs_i|^2              (512)
#define WS_N2    33280     // |X2s_j|^2             (512)
#define WS_INV2  33792     // ls^-2                 (16)
#define WS_VAR   33808     // softplus(uvar)        (1)
#define WS_KNN   33824     // kNN                   (512 x 512)

__device__ __forceinline__ float softplus_f(float x) {
    // numerically stable log1p(exp(x))
    return fmaxf(x, 0.0f) + log1pf(__expf(-fabsf(x)));
}

// ---------------------------------------------------------------------------
// Stage 1: hyperparameters + scaled copies of X / X2 + row norms.
// One block of 512 threads (one thread per point).
// ---------------------------------------------------------------------------
__global__ void prep_kernel(const float* __restrict__ X,
                            const float* __restrict__ X2,
                            const float* __restrict__ uls,
                            const float* __restrict__ uvar,
                            float* __restrict__ ws) {
    __shared__ float s_ls[DDIM];
    __shared__ float s_i2[DDIM];
    int t = threadIdx.x;
    if (t < DDIM) {
        float sp = softplus_f(uls[t]);
        s_ls[t] = sp;
        float i2 = 1.0f / (sp * sp);
        s_i2[t] = i2;
        ws[WS_INV2 + t] = i2;
    }
    if (t == DDIM) {
        ws[WS_VAR] = softplus_f(uvar[0]);
    }
    __syncthreads();

    int n = t;  // 0..511
    float accx = 0.0f, acc2 = 0.0f;
#pragma unroll
    for (int d = 0; d < DDIM; ++d) {
        float ls = s_ls[d], i2 = s_i2[d];
        float xv = X[n * DDIM + d];
        float xs = xv / ls;
        ws[WS_XS + n * DDIM + d]  = xs;
        ws[WS_PC + d * NPTS + n]  = xv * i2;
        accx = fmaf(xs, xs, accx);
        float yv = X2[n * DDIM + d];
        float ys = yv / ls;
        ws[WS_X2S + n * DDIM + d] = ys;
        ws[WS_PC2 + d * NPTS + n] = yv * i2;
        acc2 = fmaf(ys, ys, acc2);
    }
    ws[WS_NX + n] = accx;
    ws[WS_N2 + n] = acc2;
}

// ---------------------------------------------------------------------------
// Stage 2: Gram / RBF kernel via V_WMMA_F32_16X16X4_F32 (K = 16 -> 4 chained
// WMMAs). One wave (32 lanes) produces one 16x16 tile of
//     out[i,j] = var * exp(-0.25 * (na[i] - 2*A_i.B_j + nb[j])).
//
// f32 A-matrix 16x4 lane layout (ISA 7.12.2): lanes 0-15 hold M=lane with
// VGPR0=K0,VGPR1=K1; lanes 16-31 hold M=lane-16 with VGPR0=K2,VGPR1=K3.
// B 4x16 mirrors it with N=lane. C/D 16x16: VGPR r -> M=r (lanes 0-15) or
// M=r+8 (lanes 16-31), N=lane%16.
// ---------------------------------------------------------------------------
__global__ void gram_kernel(const float* __restrict__ A,   // 512 x 16 scaled
                            const float* __restrict__ B,   // 512 x 16 scaled
                            const float* __restrict__ na,  // 512
                            const float* __restrict__ nb,  // 512
                            const float* __restrict__ varp,
                            float* __restrict__ out)       // 512 x 512
{
    int ti = blockIdx.x;          // row tile
    int tj = blockIdx.y;          // col tile
    int lane = threadIdx.x;       // 0..31 (full wave, EXEC all ones)
    int half = lane >> 4;         // 0/1 -> K pair select
    int l    = lane & 15;

    int rowA = ti * 16 + l;
    int rowB = tj * 16 + l;

    v8f c = {};
#pragma unroll
    for (int kk = 0; kk < DDIM; kk += 4) {
        int k0 = kk + 2 * half;
        v2f a, b;
        a.x = A[rowA * DDIM + k0];
        a.y = A[rowA * DDIM + k0 + 1];
        b.x = B[rowB * DDIM + k0];
        b.y = B[rowB * DDIM + k0 + 1];
        c = __builtin_amdgcn_wmma_f32_16x16x4_f32(
                /*neg_a=*/false, a, /*neg_b=*/false, b,
                /*c_mod=*/(short)0, c, /*reuse_a=*/false, /*reuse_b=*/false);
    }

    float var = varp[0];
    int j = tj * 16 + l;
    float nbj = nb[j];
#pragma unroll
    for (int r = 0; r < 8; ++r) {
        int i = ti * 16 + r + 8 * half;
        float sq = na[i] - 2.0f * c[r] + nbj;
        out[i * NPTS + j] = var * __expf(-0.25f * sq);
    }
}

// ---------------------------------------------------------------------------
// Stage 3: grad_K[d*512+n, m] = -0.5 * (Pc[d,n] - Pc2[d,m]) * K[n,m]
// Streaming writer: one float4 per thread (b128 loads + b128 store).
// ---------------------------------------------------------------------------
__global__ void grad_kernel(const float* __restrict__ Pc,   // 16 x 512
                            const float* __restrict__ Pc2,  // 16 x 512
                            const float* __restrict__ K,    // 512 x 512
                            float* __restrict__ g)          // 8192 x 512
{
    int idx = blockIdx.x * blockDim.x + threadIdx.x;  // quad index
    int m0  = (idx & 127) << 2;   // 128 quads per 512-wide row
    int row = idx >> 7;           // 0..8191
    int d = row >> 9;
    int n = row & 511;

    float pn = Pc[d * NPTS + n];  // wave-uniform
    const float4 pm = *(const float4*)(Pc2 + d * NPTS + m0);
    const float4 kk = *(const float4*)(K + n * NPTS + m0);
    float4 o;
    o.x = -0.5f * (pn - pm.x) * kk.x;
    o.y = -0.5f * (pn - pm.y) * kk.y;
    o.z = -0.5f * (pn - pm.z) * kk.z;
    o.w = -0.5f * (pn - pm.w) * kk.w;
    *(float4*)(g + (size_t)row * NPTS + m0) = o;
}

// ---------------------------------------------------------------------------
// Stage 4: hess_K[a*512+i, b*512+j] = kNN[i,j]*(diag_ab - 0.25*s_a*s_b)
// with s_a = Pc[a,i]-Pc[a,j], diag_ab = (a==b) * 0.5 * ls_a^-2.
// Pure streaming writer (268 MB) -> b128 stores, L2-resident reads.
// ---------------------------------------------------------------------------
__global__ void hess_kernel(const float* __restrict__ Pc,    // 16 x 512
                            const float* __restrict__ inv2,  // 16
                            const float* __restrict__ kNN,   // 512 x 512
                            float* __restrict__ h)           // 8192 x 8192
{
    int idx = blockIdx.x * blockDim.x + threadIdx.x;  // quad index (< 2^24)
    int c0  = (idx & 2047) << 2;  // 2048 quads per 8192-wide row
    int row = idx >> 11;          // 0..8191
    int a = row >> 9;
    int i = row & 511;
    int b = c0 >> 9;
    int j = c0 & 511;

    float pai = Pc[a * NPTS + i];                       // wave-uniform
    float pbi = Pc[b * NPTS + i];                       // wave-uniform
    float dab = (a == b) ? 0.5f * inv2[a] : 0.0f;       // wave-uniform

    const float4 paj = *(const float4*)(Pc + a * NPTS + j);
    const float4 pbj = *(const float4*)(Pc + b * NPTS + j);
    const float4 k4  = *(const float4*)(kNN + i * NPTS + j);

    float4 o;
    o.x = k4.x * fmaf(pai - paj.x, -0.25f * (pbi - pbj.x), dab);
    o.y = k4.y * fmaf(pai - paj.y, -0.25f * (pbi - pbj.y), dab);
    o.z = k4.z * fmaf(pai - paj.z, -0.25f * (pbi - pbj.z), dab);
    o.w = k4.w * fmaf(pai - paj.w, -0.25f * (pbi - pbj.w), dab);
    *(float4*)(h + (size_t)row * 8192 + c0) = o;
}

// ---------------------------------------------------------------------------
extern "C" void kernel_launch(void* const* d_in, const int* in_sizes, int n_in,
                              void* d_out, int out_size, void* d_ws, size_t ws_size,
                              hipStream_t stream) {
    const float* X    = (const float*)d_in[0];
    const float* X2   = (const float*)d_in[1];
    const float* uls  = (const float*)d_in[2];
    const float* uvar = (const float*)d_in[3];

    float* out = (float*)d_out;
    float* K   = out;                                   // 262144
    float* gK  = out + NPTS * NPTS;                     // 4194304
    float* hK  = out + NPTS * NPTS + (size_t)NPTS * DDIM * NPTS;

    float* ws   = (float*)d_ws;
    float* Xs   = ws + WS_XS;
    float* X2s  = ws + WS_X2S;
    float* Pc   = ws + WS_PC;
    float* Pc2  = ws + WS_PC2;
    float* nx   = ws + WS_NX;
    float* n2   = ws + WS_N2;
    float* inv2 = ws + WS_INV2;
    float* varp = ws + WS_VAR;
    float* kNN  = ws + WS_KNN;

    // 1) hyperparams + scaled inputs
    prep_kernel<<<1, NPTS, 0, stream>>>(X, X2, uls, uvar, ws);

    // 2) RBF grams via f32 WMMA: K(X,X2) to output, kNN(X,X) to workspace
    dim3 tiles(NPTS / 16, NPTS / 16);  // 32 x 32 tiles, one wave each
    gram_kernel<<<tiles, 32, 0, stream>>>(Xs, X2s, nx, n2, varp, K);
    gram_kernel<<<tiles, 32, 0, stream>>>(Xs, Xs, nx, nx, varp, kNN);

    // 3) grad_K: 8192x512 = 1,048,576 quads
    grad_kernel<<<4096, 256, 0, stream>>>(Pc, Pc2, K, gK);

    // 4) hess_K: 8192x8192 = 16,777,216 quads (store-bandwidth bound)
    hess_kernel<<<65536, 256, 0, stream>>>(Pc, inv2, kNN, hK);
}